// MessageFunction_45045617000718
// MI455X (gfx1250) — compile-verified
//
#include <hip/hip_runtime.h>

typedef __attribute__((ext_vector_type(2))) float v2f;
typedef __attribute__((ext_vector_type(8))) float v8f;

#define NVAL 50000          // N nodes
#define KF   128            // reduction depth (EDGE_F == NODE_F)
#define HALF_M 128          // output rows per GEMM half
#define KG (KF / 4)         // 32 k-groups of depth 4
#define TILES_PER_WAVE 5    // 3125 n-tiles = 5 * 625

// One wave (32 threads) owns (gemm, batch, m-pair): TWO 16-row m-tiles.
// A-fragments for both m-tiles (128 VGPRs) are loaded once and reused for
// TILES_PER_WAVE n-tiles. Each n-tile loads B-fragments once (64 VGPRs) and
// feeds two independent 32-deep WMMA chains -> halves vmem issue and X's L2
// read amplification vs one m-tile per wave.
__global__ __launch_bounds__(32) void msg_gemm_wmma_f32(
    const float* __restrict__ h_w,   // [B, KF, N]
    const float* __restrict__ e_vw,  // [B, KF, N]
    const float* __restrict__ W_e,   // [KF, HALF_M] row-major
    const float* __restrict__ b_e,   // [HALF_M]
    const float* __restrict__ W_n,   // [KF, HALF_M]
    const float* __restrict__ b_n,   // [HALF_M]
    float* __restrict__ out)         // [B, 2*HALF_M, N]
{
  // blockIdx.x in [0,32): mpair(2 bits) | batch(2 bits) | gemm(1 bit)
  const int flat  = blockIdx.x;
  const int mpair = flat & 3;
  const int batch = (flat >> 2) & 3;
  const int gemm  = (flat >> 4) & 1;

  const float* __restrict__ X    = gemm ? h_w : e_vw;
  const float* __restrict__ W    = gemm ? W_n : W_e;
  const float* __restrict__ bias = gemm ? b_n : b_e;
  const int moff = gemm ? HALF_M : 0;

  const int lane = threadIdx.x;   // 0..31 (wave32)
  const int half = lane >> 4;     // lane-half selects K pair / +8 row group
  const int r    = lane & 15;

  const int m0 = mpair * 32;      // this wave covers rows m0 .. m0+31

  // ---- A fragments for both m-tiles: A[M][K] = W[K][M]
  // ISA f32 16x4 layout: VGPR0 = {K=0 | K=2}, VGPR1 = {K=1 | K=3} per lane-half
  v2f a0[KG], a1[KG];
#pragma unroll
  for (int kg = 0; kg < KG; ++kg) {
    const int k = kg * 4 + half * 2;
    a0[kg].x = W[(k    ) * HALF_M + m0 + r];
    a0[kg].y = W[(k + 1) * HALF_M + m0 + r];
    a1[kg].x = W[(k    ) * HALF_M + m0 + 16 + r];
    a1[kg].y = W[(k + 1) * HALF_M + m0 + 16 + r];
  }

  // Bias values for the 8 C/D rows of each tile (row = base + v + 8*half)
  float bv0[8], bv1[8];
#pragma unroll
  for (int v = 0; v < 8; ++v) {
    bv0[v] = bias[m0 + v + 8 * half];
    bv1[v] = bias[m0 + 16 + v + 8 * half];
  }

  const float* __restrict__ Xb = X + (size_t)batch * KF * NVAL;
  float* __restrict__ Ob =
      out + (size_t)batch * (2 * HALF_M) * NVAL + (size_t)(moff + m0) * NVAL;

  for (int t = 0; t < TILES_PER_WAVE; ++t) {
    const int n0 = (blockIdx.y * TILES_PER_WAVE + t) * 16;
    const float* __restrict__ Xcol = Xb + n0 + r;

    // ---- B fragments: B[K][N=n0+r]; VGPR0 = {K=0 | K=2} per lane-half
    v2f bfrag[KG];
#pragma unroll
    for (int kg = 0; kg < KG; ++kg) {
      const int k = kg * 4 + half * 2;
      bfrag[kg].x = Xcol[(size_t)(k    ) * NVAL];
      bfrag[kg].y = Xcol[(size_t)(k + 1) * NVAL];
    }

    // ---- two independent 32-deep WMMA chains over K = 128
    v8f acc0 = {}, acc1 = {};
#pragma unroll
    for (int kg = 0; kg < KG; ++kg) {
      acc0 = __builtin_amdgcn_wmma_f32_16x16x4_f32(
          false, a0[kg], false, bfrag[kg], (short)0, acc0, false, false);
      acc1 = __builtin_amdgcn_wmma_f32_16x16x4_f32(
          false, a1[kg], false, bfrag[kg], (short)0, acc1, false, false);
    }

    // ---- bias + ReLU + non-temporal store (output is write-once streaming)
    float* __restrict__ Ocol = Ob + n0 + r;
#pragma unroll
    for (int v = 0; v < 8; ++v) {
      float val0 = acc0[v] + bv0[v];
      val0 = val0 > 0.0f ? val0 : 0.0f;
      __builtin_nontemporal_store(val0, Ocol + (size_t)(v + 8 * half) * NVAL);

      float val1 = acc1[v] + bv1[v];
      val1 = val1 > 0.0f ? val1 : 0.0f;
      __builtin_nontemporal_store(val1,
                                  Ocol + (size_t)(16 + v + 8 * half) * NVAL);
    }
  }
}

extern "C" void kernel_launch(void* const* d_in, const int* in_sizes, int n_in,
                              void* d_out, int out_size, void* d_ws,
                              size_t ws_size, hipStream_t stream) {
  // setup_inputs order: 0=h_v (UNUSED), 1=h_w, 2=e_vw, 3=W_e, 4=b_e, 5=W_n, 6=b_n
  const float* h_w  = (const float*)d_in[1];
  const float* e_vw = (const float*)d_in[2];
  const float* W_e  = (const float*)d_in[3];
  const float* b_e  = (const float*)d_in[4];
  const float* W_n  = (const float*)d_in[5];
  const float* b_n  = (const float*)d_in[6];
  float* out = (float*)d_out;

  // 32 (gemm x batch x m-pair) combos, 625 n-chunks of 5 tiles each
  dim3 grid(32, (NVAL / 16) / TILES_PER_WAVE);
  dim3 block(32);
  msg_gemm_wmma_f32<<<grid, block, 0, stream>>>(h_w, e_vw, W_e, b_e, W_n, b_n,
                                                out);
}